// GAE_GConvLSTM_decoder_6794638262631
// MI455X (gfx1250) — compile-verified
//
#include <hip/hip_runtime.h>
#include <math.h>

// Problem constants (fixed by the reference module)
#define NC_    50000
#define NF_    200000
#define EC_    1600000
#define EF_    3200000
#define TSTEPS 8

typedef __attribute__((ext_vector_type(2))) float v2f;
typedef __attribute__((ext_vector_type(8))) float v8f;

__device__ __forceinline__ float elu_f(float x) { return x > 0.f ? x : (expf(x) - 1.f); }
__device__ __forceinline__ float sig_f(float x) { return 1.f / (1.f + expf(-x)); }

// ---------------------------------------------------------------- utilities
__global__ void fill_f32(float* __restrict__ p, float v, int n) {
  int i = blockIdx.x * blockDim.x + threadIdx.x;
  if (i < n) p[i] = v;
}

__global__ void deg_accum(const int* __restrict__ dst, const float* __restrict__ ew,
                          float* __restrict__ deg, int E) {
  int e = blockIdx.x * blockDim.x + threadIdx.x;
  if (e < E)
    __hip_atomic_fetch_add(&deg[dst[e]], ew[e], __ATOMIC_RELAXED, __HIP_MEMORY_SCOPE_AGENT);
}

__global__ void rsqrt_ip(float* __restrict__ d, int n) {
  int i = blockIdx.x * blockDim.x + threadIdx.x;
  if (i < n) d[i] = rsqrtf(d[i]);
}

// ------------------------------------------------------- f32 WMMA GEMM core
// C[M,N] = act( A[M,K](lda) @ W[K,N](ldw) + bias )  ; one wave per 16x16 tile
// grid = (M/16, N/16), block = 32 (single wave, EXEC all-1s, no divergence)
__global__ void wmma_gemm(const float* __restrict__ A, int lda,
                          const float* __restrict__ W, int ldw,
                          const float* __restrict__ bias,
                          float* __restrict__ C, int ldc,
                          int K, int act) {
  const int lane = threadIdx.x;
  const int half = lane >> 4;     // 0: K+0..1 / rows M+0..7 ; 1: K+2..3 / rows M+8..15
  const int l15  = lane & 15;
  const int arow = blockIdx.x * 16 + l15;   // A row for this lane
  const int ncol = blockIdx.y * 16 + l15;   // C/B column for this lane

  v8f acc = {};
  const float* ap = A + (size_t)arow * lda + half * 2;          // A: 16x4 tile operand
  const float* wp = W + (size_t)(half * 2) * ldw + ncol;        // B: 4x16 tile operand
  for (int k0 = 0; k0 < K; k0 += 4) {
    v2f a, b;
    a.x = ap[0];   a.y = ap[1];          // K = k0+2*half, k0+2*half+1
    b.x = wp[0];   b.y = wp[ldw];        // rows K = k0+2*half, +1 ; col = ncol
    ap += 4;
    wp += (size_t)4 * ldw;
    acc = __builtin_amdgcn_wmma_f32_16x16x4_f32(false, a, false, b,
                                                (short)0, acc, false, false);
  }

  const float bv = bias ? bias[ncol] : 0.f;
  const int rbase = blockIdx.x * 16 + half * 8;   // C/D: VGPR v -> row v (+8 for hi half)
#pragma unroll
  for (int v = 0; v < 8; ++v) {
    float r = acc[v] + bv;
    if (act == 1) r = elu_f(r);
    C[(size_t)(rbase + v) * ldc + ncol] = r;
  }
}

// ------------------------------------------------------------- GCN scatter
// agg[dst] += dinv[src]*ew*dinv[dst] * xw[src] ; one wave per edge, lanes = features
__global__ void gcn_scatter(const int* __restrict__ src, const int* __restrict__ dst,
                            const float* __restrict__ ew, const float* __restrict__ dinv,
                            const float* __restrict__ xw, float* __restrict__ agg,
                            int E, int F) {
  int wid  = (blockIdx.x * blockDim.x + threadIdx.x) >> 5;
  int lane = threadIdx.x & 31;
  if (wid >= E) return;
  int s = src[wid], d = dst[wid];
  float coef = dinv[s] * ew[wid] * dinv[d];
  const float* xs = xw + (size_t)s * F;
  float* ad = agg + (size_t)d * F;
  for (int f = lane; f < F; f += 32)
    __hip_atomic_fetch_add(&ad[f], coef * xs[f], __ATOMIC_RELAXED, __HIP_MEMORY_SCOPE_AGENT);
}

// y[node, f (ldy)] = act( agg + dinv^2 * xw + bias )
__global__ void gcn_epilogue(const float* __restrict__ agg, const float* __restrict__ xw,
                             const float* __restrict__ dinv, const float* __restrict__ bias,
                             float* __restrict__ y, int n, int F, int ldy, int act) {
  int idx = blockIdx.x * blockDim.x + threadIdx.x;
  if (idx >= n * F) return;
  int node = idx / F, f = idx - node * F;
  float di = dinv[node];
  float r = agg[idx] + di * di * xw[idx] + (bias ? bias[f] : 0.f);
  if (act == 1) r = elu_f(r);
  y[(size_t)node * ldy + f] = r;
}

// ----------------------------------------------------------- fused helpers
__global__ void concat_hc(const float* __restrict__ h, const float* __restrict__ c,
                          float* __restrict__ cat, int n) {
  int idx = blockIdx.x * blockDim.x + threadIdx.x;
  if (idx >= n * 64) return;
  int node = idx >> 6, f = idx & 63;
  cat[idx] = (f < 32) ? h[node * 32 + f] : c[node * 32 + (f - 32)];
}

__global__ void init_state(const float* __restrict__ agg, const float* __restrict__ xw,
                           const float* __restrict__ dinv, const float* __restrict__ bias,
                           float* __restrict__ ht, float* __restrict__ ct, int n) {
  int idx = blockIdx.x * blockDim.x + threadIdx.x;
  if (idx >= n * 64) return;
  int node = idx >> 6, f = idx & 63;
  float di = dinv[node];
  float r = elu_f(agg[idx] + di * di * xw[idx] + bias[f]);
  if (f < 32) ht[node * 32 + f] = r;
  else        ct[node * 32 + (f - 32)] = r;
}

__global__ void pack_w4(const float* __restrict__ Wi, const float* __restrict__ Wf,
                        const float* __restrict__ Wg, const float* __restrict__ Wo,
                        float* __restrict__ W4) {
  int idx = blockIdx.x * blockDim.x + threadIdx.x;   // 32*128
  if (idx >= 32 * 128) return;
  int k = idx >> 7, col = idx & 127;
  int g = col >> 5, f = col & 31;
  const float* W = (g == 0) ? Wi : (g == 1) ? Wf : (g == 2) ? Wg : Wo;
  W4[idx] = W[k * 32 + f];
}

__global__ void pack_b4(const float* __restrict__ bi, const float* __restrict__ bf,
                        const float* __restrict__ bg, const float* __restrict__ bo,
                        float* __restrict__ b4) {
  int col = blockIdx.x * blockDim.x + threadIdx.x;
  if (col >= 128) return;
  int g = col >> 5, f = col & 31;
  const float* b = (g == 0) ? bi : (g == 1) ? bf : (g == 2) ? bg : bo;
  b4[col] = b[f];
}

// gates: pre = xgate4 + hagg4 + dinv^2*hw4 ; LSTM cell update in place
__global__ void lstm_gate(const float* __restrict__ xg4, const float* __restrict__ hagg4,
                          const float* __restrict__ hw4, const float* __restrict__ dinv,
                          float* __restrict__ ht, float* __restrict__ ct, int n) {
  int idx = blockIdx.x * blockDim.x + threadIdx.x;   // n*32
  if (idx >= n * 32) return;
  int node = idx >> 5, j = idx & 31;
  float di = dinv[node], d2 = di * di;
  size_t b = (size_t)node * 128;
  float pi = xg4[b + j]      + hagg4[b + j]      + d2 * hw4[b + j];
  float pf = xg4[b + 32 + j] + hagg4[b + 32 + j] + d2 * hw4[b + 32 + j];
  float pg = xg4[b + 64 + j] + hagg4[b + 64 + j] + d2 * hw4[b + 64 + j];
  float po = xg4[b + 96 + j] + hagg4[b + 96 + j] + d2 * hw4[b + 96 + j];
  float iv = sig_f(pi), fv = sig_f(pf), gv = tanhf(pg), ov = sig_f(po);
  float cn = fv * ct[idx] + iv * gv;
  ct[idx] = cn;
  ht[idx] = ov * tanhf(cn);
}

__global__ void gather32(const float* __restrict__ x, const int* __restrict__ cl,
                         float* __restrict__ xf, int nf) {
  int idx = blockIdx.x * blockDim.x + threadIdx.x;   // nf*32
  if (idx >= nf * 32) return;
  int i = idx >> 5, j = idx & 31;
  xf[idx] = x[(size_t)cl[i] * 32 + j];
}

// constant columns 32..39 of the padded m1 input (pos, node_attrs, zero pad)
__global__ void fill_m1_extra(const float* __restrict__ pos, const float* __restrict__ na,
                              float* __restrict__ xfin, int nf) {
  int idx = blockIdx.x * blockDim.x + threadIdx.x;   // nf*8
  if (idx >= nf * 8) return;
  int i = idx >> 3, j = idx & 7;
  float v = (j < 3) ? pos[i * 3 + j] : (j < 6) ? na[i * 3 + (j - 3)] : 0.f;
  xfin[(size_t)i * 40 + 32 + j] = v;
}

__global__ void pad_wm1(const float* __restrict__ Wm1, float* __restrict__ Wp) {
  int idx = blockIdx.x * blockDim.x + threadIdx.x;   // 40*32
  if (idx >= 40 * 32) return;
  int k = idx >> 5, f = idx & 31;
  Wp[idx] = (k < 38) ? Wm1[k * 32 + f] : 0.f;
}

// out = z @ W_m2 (32x3) + b_m2
__global__ void out_proj(const float* __restrict__ z, const float* __restrict__ W,
                         const float* __restrict__ b, float* __restrict__ out, int nf) {
  int idx = blockIdx.x * blockDim.x + threadIdx.x;   // nf*3
  if (idx >= nf * 3) return;
  int i = idx / 3, o = idx - i * 3;
  const float* zr = z + (size_t)i * 32;
  float acc = b[o];
#pragma unroll
  for (int k = 0; k < 32; ++k) acc += zr[k] * W[k * 3 + o];
  out[idx] = acc;
}

// ---------------------------------------------------------------- launcher
static inline unsigned nblk(long n) { return (unsigned)((n + 255) / 256); }

extern "C" void kernel_launch(void* const* d_in, const int* in_sizes, int n_in,
                              void* d_out, int out_size, void* d_ws, size_t ws_size,
                              hipStream_t stream) {
  (void)in_sizes; (void)n_in; (void)out_size; (void)ws_size;
  const float* h      = (const float*)d_in[0];
  const float* c      = (const float*)d_in[1];
  /* d_in[2] seq_len: fixed T=8 in this module */
  const int*   hei    = (const int*)d_in[3];
  const float* hea    = (const float*)d_in[4];
  const int*   ei     = (const int*)d_in[5];
  const float* ea     = (const float*)d_in[6];
  /* d_in[7] edge_indices_f2c: unused by the reference */
  const float* pos    = (const float*)d_in[8];
  const float* na     = (const float*)d_in[9];
  const int*   cl     = (const int*)d_in[10];
  const float* W_init = (const float*)d_in[11];
  const float* b_init = (const float*)d_in[12];
  const float* W_xi = (const float*)d_in[13], *W_hi = (const float*)d_in[14], *b_i = (const float*)d_in[15];
  const float* W_xf = (const float*)d_in[16], *W_hf = (const float*)d_in[17], *b_f = (const float*)d_in[18];
  const float* W_xg = (const float*)d_in[19], *W_hg = (const float*)d_in[20], *b_g = (const float*)d_in[21];
  const float* W_xo = (const float*)d_in[22], *W_ho = (const float*)d_in[23], *b_o = (const float*)d_in[24];
  const float* W_mp1 = (const float*)d_in[25], *b_mp1 = (const float*)d_in[26];
  const float* W_mp2 = (const float*)d_in[27], *b_mp2 = (const float*)d_in[28];
  const float* W_fine = (const float*)d_in[29], *b_fine = (const float*)d_in[30];
  const float* W_m1 = (const float*)d_in[31], *b_m1 = (const float*)d_in[32];
  const float* W_m2 = (const float*)d_in[33], *b_m2 = (const float*)d_in[34];
  float* out = (float*)d_out;

  const int* src  = hei;       const int* dstv = hei + EC_;
  const int* fsrc = ei;        const int* fdst = ei + EF_;

  // workspace carve-out (all f32)
  float* ws = (float*)d_ws;
  size_t off = 0;
  auto alloc = [&](size_t nel) { float* p = ws + off; off += nel; return p; };
  float* dinv_c = alloc(NC_);
  float* dinv_f = alloc(NF_);
  float* cat    = alloc((size_t)NC_ * 64);
  float* xw64   = alloc((size_t)NC_ * 64);
  float* agg64  = alloc((size_t)NC_ * 64);
  float* ht     = alloc((size_t)NC_ * 32);
  float* ct     = alloc((size_t)NC_ * 32);
  float* W4x    = alloc(32 * 128);
  float* W4h    = alloc(32 * 128);
  float* b4     = alloc(128);
  float* xw4    = alloc((size_t)NC_ * 128);
  float* xg4    = alloc((size_t)NC_ * 128);
  float* hw4    = alloc((size_t)NC_ * 128);
  float* hagg4  = alloc((size_t)NC_ * 128);
  float* xw32   = alloc((size_t)NC_ * 32);
  float* agg32  = alloc((size_t)NC_ * 32);
  float* x1     = alloc((size_t)NC_ * 32);
  float* x2     = alloc((size_t)NC_ * 32);
  float* xf     = alloc((size_t)NF_ * 32);
  float* xfw    = alloc((size_t)NF_ * 32);
  float* fagg   = alloc((size_t)NF_ * 32);
  float* xfin   = alloc((size_t)NF_ * 40);
  float* Wm1p   = alloc(40 * 32);
  float* z      = alloc((size_t)NF_ * 32);

  // ---- degree / normalization (once per graph)
  fill_f32<<<nblk(NC_), 256, 0, stream>>>(dinv_c, 1.0f, NC_);
  deg_accum<<<nblk(EC_), 256, 0, stream>>>(dstv, hea, dinv_c, EC_);
  rsqrt_ip<<<nblk(NC_), 256, 0, stream>>>(dinv_c, NC_);
  fill_f32<<<nblk(NF_), 256, 0, stream>>>(dinv_f, 1.0f, NF_);
  deg_accum<<<nblk(EF_), 256, 0, stream>>>(fdst, ea, dinv_f, EF_);
  rsqrt_ip<<<nblk(NF_), 256, 0, stream>>>(dinv_f, NF_);

  // ---- initial state: states = elu(gcn([h|c], W_init, b_init))
  concat_hc<<<nblk((long)NC_ * 64), 256, 0, stream>>>(h, c, cat, NC_);
  wmma_gemm<<<dim3(NC_ / 16, 64 / 16), 32, 0, stream>>>(cat, 64, W_init, 64, nullptr,
                                                        xw64, 64, 64, 0);
  fill_f32<<<nblk((long)NC_ * 64), 256, 0, stream>>>(agg64, 0.f, NC_ * 64);
  gcn_scatter<<<nblk((long)EC_ * 32), 256, 0, stream>>>(src, dstv, hea, dinv_c, xw64, agg64, EC_, 64);
  init_state<<<nblk((long)NC_ * 64), 256, 0, stream>>>(agg64, xw64, dinv_c, b_init, ht, ct, NC_);

  // ---- pack fused gate weights; precompute x-gate GCNs once (h is constant)
  pack_w4<<<nblk(32 * 128), 256, 0, stream>>>(W_xi, W_xf, W_xg, W_xo, W4x);
  pack_w4<<<nblk(32 * 128), 256, 0, stream>>>(W_hi, W_hf, W_hg, W_ho, W4h);
  pack_b4<<<1, 128, 0, stream>>>(b_i, b_f, b_g, b_o, b4);
  wmma_gemm<<<dim3(NC_ / 16, 128 / 16), 32, 0, stream>>>(h, 32, W4x, 128, nullptr,
                                                         xw4, 128, 32, 0);
  fill_f32<<<nblk((long)NC_ * 128), 256, 0, stream>>>(hagg4, 0.f, NC_ * 128);
  gcn_scatter<<<nblk((long)EC_ * 32), 256, 0, stream>>>(src, dstv, hea, dinv_c, xw4, hagg4, EC_, 128);
  gcn_epilogue<<<nblk((long)NC_ * 128), 256, 0, stream>>>(hagg4, xw4, dinv_c, b4,
                                                          xg4, NC_, 128, 128, 0);

  // ---- constant columns of m1 input; padded W_m1
  fill_m1_extra<<<nblk((long)NF_ * 8), 256, 0, stream>>>(pos, na, xfin, NF_);
  pad_wm1<<<nblk(40 * 32), 256, 0, stream>>>(W_m1, Wm1p);

  // ---- time loop
  for (int t = 0; t < TSTEPS; ++t) {
    // fused h-gate GCNs: one GEMM + one 128-feature scatter
    wmma_gemm<<<dim3(NC_ / 16, 128 / 16), 32, 0, stream>>>(ht, 32, W4h, 128, nullptr,
                                                           hw4, 128, 32, 0);
    fill_f32<<<nblk((long)NC_ * 128), 256, 0, stream>>>(hagg4, 0.f, NC_ * 128);
    gcn_scatter<<<nblk((long)EC_ * 32), 256, 0, stream>>>(src, dstv, hea, dinv_c, hw4, hagg4, EC_, 128);
    lstm_gate<<<nblk((long)NC_ * 32), 256, 0, stream>>>(xg4, hagg4, hw4, dinv_c, ht, ct, NC_);

    // x = elu(gcn(h_new, W_mp1, b_mp1))
    wmma_gemm<<<dim3(NC_ / 16, 32 / 16), 32, 0, stream>>>(ht, 32, W_mp1, 32, nullptr,
                                                          xw32, 32, 32, 0);
    fill_f32<<<nblk((long)NC_ * 32), 256, 0, stream>>>(agg32, 0.f, NC_ * 32);
    gcn_scatter<<<nblk((long)EC_ * 32), 256, 0, stream>>>(src, dstv, hea, dinv_c, xw32, agg32, EC_, 32);
    gcn_epilogue<<<nblk((long)NC_ * 32), 256, 0, stream>>>(agg32, xw32, dinv_c, b_mp1,
                                                           x1, NC_, 32, 32, 1);
    // x = elu(gcn(x, W_mp2, b_mp2))
    wmma_gemm<<<dim3(NC_ / 16, 32 / 16), 32, 0, stream>>>(x1, 32, W_mp2, 32, nullptr,
                                                          xw32, 32, 32, 0);
    fill_f32<<<nblk((long)NC_ * 32), 256, 0, stream>>>(agg32, 0.f, NC_ * 32);
    gcn_scatter<<<nblk((long)EC_ * 32), 256, 0, stream>>>(src, dstv, hea, dinv_c, xw32, agg32, EC_, 32);
    gcn_epilogue<<<nblk((long)NC_ * 32), 256, 0, stream>>>(agg32, xw32, dinv_c, b_mp2,
                                                           x2, NC_, 32, 32, 1);

    // fine level: gather + gcn(xf, W_fine, b_fine) over EF edges
    gather32<<<nblk((long)NF_ * 32), 256, 0, stream>>>(x2, cl, xf, NF_);
    wmma_gemm<<<dim3(NF_ / 16, 32 / 16), 32, 0, stream>>>(xf, 32, W_fine, 32, nullptr,
                                                          xfw, 32, 32, 0);
    fill_f32<<<nblk((long)NF_ * 32), 256, 0, stream>>>(fagg, 0.f, NF_ * 32);
    gcn_scatter<<<nblk((long)EF_ * 32), 256, 0, stream>>>(fsrc, fdst, ea, dinv_f, xfw, fagg, EF_, 32);
    gcn_epilogue<<<nblk((long)NF_ * 32), 256, 0, stream>>>(fagg, xfw, dinv_f, b_fine,
                                                           xfin, NF_, 32, 40, 1);   // cols 0..31

    // z = elu([xf|pos|attr|0] @ Wm1p + b_m1)   (K=40 padded)
    wmma_gemm<<<dim3(NF_ / 16, 32 / 16), 32, 0, stream>>>(xfin, 40, Wm1p, 32, b_m1,
                                                          z, 32, 40, 1);
    // out[t] = z @ W_m2 + b_m2
    out_proj<<<nblk((long)NF_ * 3), 256, 0, stream>>>(z, W_m2, b_m2,
                                                      out + (size_t)t * NF_ * 3, NF_);
  }
}